// NoseHooverChains_10419590660406
// MI455X (gfx1250) — compile-verified
//
#include <hip/hip_runtime.h>
#include <math.h>

#define NATOMS 768
#define FEAT   128
#define NRBF   32
#define NTILES (NATOMS / 16)        // 48 i-tiles
#define JCHUNKS 8
#define JT_PER_CHUNK (NTILES / JCHUNKS)  // 6 j-tiles per block

static constexpr float CUT2    = 25.0f;                 // CUTOFF^2
static constexpr float GAMMA_C = 40.96f;                // 1/(CUTOFF/N_RBF)^2
static constexpr float CSTEP   = 5.0f / 31.0f;          // linspace(0,5,32) step
static constexpr float KB_C    = 8.617330337217213e-05f;
static constexpr float KT_C    = 300.0f * KB_C;
static constexpr float TT2     = 400.0f;                // TTIME^2

typedef float v2f __attribute__((ext_vector_type(2)));
typedef float v8f __attribute__((ext_vector_type(8)));

// ---- WMMA f32 16x16x4 fragment loaders (CDNA5 ISA 7.12.2 layouts) ----
// A (16x4, row-major source, leading dim ld): lane -> row = lane&15, K pair = (lane>>4)*2
__device__ __forceinline__ v2f lda(const float* s, int ld, int lane) {
  int row = lane & 15, kb = (lane >> 4) << 1;
  v2f a; a.x = s[row * ld + kb]; a.y = s[row * ld + kb + 1]; return a;
}
// B (4x16, row-major source): lane -> col = lane&15, K pair = (lane>>4)*2
__device__ __forceinline__ v2f ldb(const float* s, int ld, int lane) {
  int col = lane & 15, kb = (lane >> 4) << 1;
  v2f b; b.x = s[kb * ld + col]; b.y = s[(kb + 1) * ld + col]; return b;
}
// B = W^T fragment: B[kk][n] = W[n0+n][k0+kk]; s = &W[n0*ld + k0]
__device__ __forceinline__ v2f ldbt(const float* s, int ld, int lane) {
  int col = lane & 15, kb = (lane >> 4) << 1;
  v2f b; b.x = s[col * ld + kb]; b.y = s[col * ld + kb + 1]; return b;
}
__device__ __forceinline__ v8f wmma4(v2f a, v2f b, v8f c) {
  return __builtin_amdgcn_wmma_f32_16x16x4_f32(false, a, false, b, (short)0, c, false, false);
}

// ---------------- Kernel A: gather h0 = embed[an], zero m and f ----------------
__global__ void init_kernel(const int* __restrict__ an, const float* __restrict__ embed,
                            float* __restrict__ h0, float* __restrict__ m, float* __restrict__ f) {
  int idx = blockIdx.x * blockDim.x + threadIdx.x;
  if (idx < NATOMS * FEAT) {
    int i = idx >> 7, t = idx & 127;
    h0[idx] = embed[an[i] * FEAT + t];
    m[idx] = 0.0f;
  }
  if (idx < NATOMS * 3) f[idx] = 0.0f;
}

// ---------------- Kernel B: forward messages m_i += sum_j filt_ij * h0_j ----------------
// 16 waves; wave w owns pair rows [16w, 16w+16) (i.e. atom ii = w of the i-tile).
__global__ void __launch_bounds__(512)
fwd_msg_kernel(const float* __restrict__ q, const float* __restrict__ h0,
               const float* __restrict__ W1, const float* __restrict__ b1,
               const float* __restrict__ W2, const float* __restrict__ b2,
               float* __restrict__ mOut) {
  __shared__ float W1s[NRBF * FEAT];          // 16 KB
  __shared__ float W2s[FEAT * FEAT];          // 64 KB
  __shared__ float b1s[FEAT], b2s[FEAT];
  __shared__ float h0j[16 * FEAT];            // 8 KB
  __shared__ float rbfs[256 * NRBF];          // 32 KB
  __shared__ float t1s[256 * FEAT];           // 128 KB
  __shared__ float mAcc[16 * FEAT];           // 8 KB
  __shared__ float qis[48], qjs[48];
  __shared__ float dls[256], mls[256];

  const int tid = threadIdx.x, lane = tid & 31, w = tid >> 5;
  const int ib = blockIdx.x * 16;

  for (int idx = tid; idx < NRBF * FEAT; idx += 512) W1s[idx] = W1[idx];
  for (int idx = tid; idx < FEAT * FEAT; idx += 512) W2s[idx] = W2[idx];
  if (tid < FEAT) { b1s[tid] = b1[tid]; b2s[tid] = b2[tid]; }
  for (int idx = tid; idx < 48; idx += 512) qis[idx] = q[ib * 3 + idx];
  for (int idx = tid; idx < 16 * FEAT; idx += 512) mAcc[idx] = 0.0f;
  __syncthreads();

  for (int jt = blockIdx.y * JT_PER_CHUNK; jt < (int)(blockIdx.y + 1) * JT_PER_CHUNK; ++jt) {
    const int jb = jt * 16;
    __syncthreads();
    for (int idx = tid; idx < 16 * FEAT; idx += 512) h0j[idx] = h0[jb * FEAT + idx];
    for (int idx = tid; idx < 48; idx += 512) qjs[idx] = q[jb * 3 + idx];
    __syncthreads();
    // geometry + rbf for this wave's rows
    if (lane < 16) {
      int jj = lane, p = w * 16 + jj;
      float dx = qis[w * 3] - qjs[jj * 3], dy = qis[w * 3 + 1] - qjs[jj * 3 + 1],
            dz = qis[w * 3 + 2] - qjs[jj * 3 + 2];
      float d2 = dx * dx + dy * dy + dz * dz;
      bool msk = (d2 < CUT2) && ((ib + w) != (jb + jj));
      dls[p] = sqrtf(msk ? d2 : 1.0f);
      mls[p] = msk ? 1.0f : 0.0f;
    }
    for (int t = lane; t < 16 * NRBF; t += 32) {
      int jj = t >> 5, k = t & 31, p = w * 16 + jj;
      float dd = dls[p] - CSTEP * (float)k;
      rbfs[p * NRBF + k] = __expf(-GAMMA_C * dd * dd);
    }
    __syncthreads();
    // GEMM1: t1 = relu(rbf @ W1 + b1)   [16 x 128] rows owned by wave w
    for (int nt = 0; nt < 8; ++nt) {
      float bv = b1s[nt * 16 + (lane & 15)];
      v8f acc = {bv, bv, bv, bv, bv, bv, bv, bv};
#pragma unroll
      for (int kt = 0; kt < 8; ++kt) {
        v2f a = lda(&rbfs[(w * 16) * NRBF + kt * 4], NRBF, lane);
        v2f b = ldb(&W1s[(kt * 4) * FEAT + nt * 16], FEAT, lane);
        acc = wmma4(a, b, acc);
      }
      int hi = lane >> 4, col = lane & 15;
#pragma unroll
      for (int r = 0; r < 8; ++r)
        t1s[(w * 16 + r + 8 * hi) * FEAT + nt * 16 + col] = fmaxf(acc[r], 0.0f);
    }
    __syncthreads();
    // GEMM2: filt = t1 @ W2 + b2; masked accumulate m_i[f] += filt * h0_j[f]
    for (int nt = 0; nt < 8; ++nt) {
      float bv = b2s[nt * 16 + (lane & 15)];
      v8f acc = {bv, bv, bv, bv, bv, bv, bv, bv};
#pragma unroll 4
      for (int kt = 0; kt < 32; ++kt) {
        v2f a = lda(&t1s[(w * 16) * FEAT + kt * 4], FEAT, lane);
        v2f b = ldb(&W2s[(kt * 4) * FEAT + nt * 16], FEAT, lane);
        acc = wmma4(a, b, acc);
      }
      int hi = lane >> 4, col = lane & 15, fcol = nt * 16 + col;
      float partial = 0.0f;
#pragma unroll
      for (int r = 0; r < 8; ++r) {
        int jj = r + 8 * hi;
        partial += mls[w * 16 + jj] * acc[r] * h0j[jj * FEAT + fcol];
      }
      atomicAdd(&mAcc[w * FEAT + fcol], partial);
    }
  }
  __syncthreads();
  for (int idx = tid; idx < 16 * FEAT; idx += 512)
    atomicAdd(&mOut[ib * FEAT + idx], mAcc[idx]);
}

// ---------------- Kernel C: per-atom head  u_i = W3 (step(m_i W3) * w_out) ----------------
__global__ void peratom_u_kernel(const float* __restrict__ m, const float* __restrict__ W3,
                                 const float* __restrict__ w_out, float* __restrict__ u) {
  __shared__ float mi[FEAT];
  __shared__ float g[FEAT];
  const int i = blockIdx.x, t = threadIdx.x;
  mi[t] = m[i * FEAT + t];
  __syncthreads();
  float s = 0.0f;
  for (int f = 0; f < FEAT; ++f) s += mi[f] * W3[f * FEAT + t];
  g[t] = (s > 0.0f) ? w_out[t] : 0.0f;
  __syncthreads();
  float uv = 0.0f;
  for (int fp = 0; fp < FEAT; ++fp) uv += W3[t * FEAT + fp] * g[fp];
  u[i * FEAT + t] = uv;
}

// ---------------- Kernel D: backward pair pass -> forces ----------------
__global__ void __launch_bounds__(512)
bwd_force_kernel(const float* __restrict__ q, const float* __restrict__ h0,
                 const float* __restrict__ u, const float* __restrict__ W1,
                 const float* __restrict__ b1, const float* __restrict__ W2,
                 float* __restrict__ fOut) {
  __shared__ float W1s[NRBF * FEAT];          // 16 KB
  __shared__ float W2s[FEAT * FEAT];          // 64 KB
  __shared__ float b1s[FEAT];
  __shared__ float h0i[16 * FEAT], uis[16 * FEAT];   // 16 KB
  __shared__ float h0j[16 * FEAT], ujs[16 * FEAT];   // 16 KB
  __shared__ float rbfs[256 * NRBF];          // 32 KB
  __shared__ float cAg[256 * FEAT];           // 128 KB (c, then gated-A)
  __shared__ float SAcc[256];
  __shared__ float fAcc[48];
  __shared__ float qis[48], qjs[48];
  __shared__ float dls[256], mls[256];

  const int tid = threadIdx.x, lane = tid & 31, w = tid >> 5;
  const int ib = blockIdx.x * 16;

  for (int idx = tid; idx < NRBF * FEAT; idx += 512) W1s[idx] = W1[idx];
  for (int idx = tid; idx < FEAT * FEAT; idx += 512) W2s[idx] = W2[idx];
  if (tid < FEAT) b1s[tid] = b1[tid];
  for (int idx = tid; idx < 16 * FEAT; idx += 512) {
    h0i[idx] = h0[ib * FEAT + idx];
    uis[idx] = u[ib * FEAT + idx];
  }
  for (int idx = tid; idx < 48; idx += 512) { qis[idx] = q[ib * 3 + idx]; fAcc[idx] = 0.0f; }
  __syncthreads();

  for (int jt = blockIdx.y * JT_PER_CHUNK; jt < (int)(blockIdx.y + 1) * JT_PER_CHUNK; ++jt) {
    const int jb = jt * 16;
    __syncthreads();
    for (int idx = tid; idx < 16 * FEAT; idx += 512) {
      h0j[idx] = h0[jb * FEAT + idx];
      ujs[idx] = u[jb * FEAT + idx];
    }
    for (int idx = tid; idx < 48; idx += 512) qjs[idx] = q[jb * 3 + idx];
    __syncthreads();
    if (lane < 16) {
      int jj = lane, p = w * 16 + jj;
      float dx = qis[w * 3] - qjs[jj * 3], dy = qis[w * 3 + 1] - qjs[jj * 3 + 1],
            dz = qis[w * 3 + 2] - qjs[jj * 3 + 2];
      float d2 = dx * dx + dy * dy + dz * dz;
      bool msk = (d2 < CUT2) && ((ib + w) != (jb + jj));
      dls[p] = sqrtf(msk ? d2 : 1.0f);
      mls[p] = msk ? 1.0f : 0.0f;
      SAcc[p] = 0.0f;
    }
    for (int t = lane; t < 16 * NRBF; t += 32) {
      int jj = t >> 5, k = t & 31, p = w * 16 + jj;
      float dd = dls[p] - CSTEP * (float)k;
      rbfs[p * NRBF + k] = __expf(-GAMMA_C * dd * dd);
    }
    __syncthreads();
    // z-GEMM -> gate bits (step(z)), 8 tiles x 8 rows packed in 64 bits per lane
    unsigned long long gbits = 0ull;
#pragma unroll
    for (int nt = 0; nt < 8; ++nt) {
      float bv = b1s[nt * 16 + (lane & 15)];
      v8f acc = {bv, bv, bv, bv, bv, bv, bv, bv};
#pragma unroll
      for (int kt = 0; kt < 8; ++kt) {
        v2f a = lda(&rbfs[(w * 16) * NRBF + kt * 4], NRBF, lane);
        v2f b = ldb(&W1s[(kt * 4) * FEAT + nt * 16], FEAT, lane);
        acc = wmma4(a, b, acc);
      }
#pragma unroll
      for (int r = 0; r < 8; ++r)
        if (acc[r] > 0.0f) gbits |= 1ull << (nt * 8 + r);
    }
    // c = u_i*h0_j + u_j*h0_i for this wave's rows
    for (int t = lane; t < 16 * FEAT; t += 32) {
      int jj = t >> 7, f = t & 127;
      cAg[(w * 16 + jj) * FEAT + f] =
          uis[w * FEAT + f] * h0j[jj * FEAT + f] + ujs[jj * FEAT + f] * h0i[w * FEAT + f];
    }
    __syncthreads();
    // A-GEMM: A = c @ W2^T ; deferred gated write back into cAg
    v8f aacc[8];
#pragma unroll
    for (int nt = 0; nt < 8; ++nt) {
      v8f acc = {0.f, 0.f, 0.f, 0.f, 0.f, 0.f, 0.f, 0.f};
#pragma unroll 4
      for (int kt = 0; kt < 32; ++kt) {
        v2f a = lda(&cAg[(w * 16) * FEAT + kt * 4], FEAT, lane);
        v2f b = ldbt(&W2s[(nt * 16) * FEAT + kt * 4], FEAT, lane);
        acc = wmma4(a, b, acc);
      }
      aacc[nt] = acc;
    }
    __syncthreads();
    {
      int hi = lane >> 4, col = lane & 15;
#pragma unroll
      for (int nt = 0; nt < 8; ++nt)
#pragma unroll
        for (int r = 0; r < 8; ++r)
          cAg[(w * 16 + r + 8 * hi) * FEAT + nt * 16 + col] =
              ((gbits >> (nt * 8 + r)) & 1ull) ? aacc[nt][r] : 0.0f;
    }
    __syncthreads();
    // B-GEMM: Bk = Ag @ W1^T ; fold with d(rbf)/dd into pair scalar S
#pragma unroll
    for (int ct = 0; ct < 2; ++ct) {
      v8f acc = {0.f, 0.f, 0.f, 0.f, 0.f, 0.f, 0.f, 0.f};
#pragma unroll 4
      for (int kt = 0; kt < 32; ++kt) {
        v2f a = lda(&cAg[(w * 16) * FEAT + kt * 4], FEAT, lane);
        v2f b = ldbt(&W1s[(ct * 16) * FEAT + kt * 4], FEAT, lane);
        acc = wmma4(a, b, acc);
      }
      int hi = lane >> 4, k = ct * 16 + (lane & 15);
      float ck = CSTEP * (float)k;
#pragma unroll
      for (int r = 0; r < 8; ++r) {
        int p = w * 16 + r + 8 * hi;
        float d = dls[p];
        float drb = -2.0f * GAMMA_C * (d - ck) * rbfs[p * NRBF + k];
        atomicAdd(&SAcc[p], acc[r] * drb * mls[p]);
      }
    }
    __syncthreads();
    // f_i -= S * disp / d
    if (lane < 16) {
      int jj = lane, p = w * 16 + jj;
      float s = -SAcc[p] / dls[p];
      atomicAdd(&fAcc[w * 3 + 0], s * (qis[w * 3 + 0] - qjs[jj * 3 + 0]));
      atomicAdd(&fAcc[w * 3 + 1], s * (qis[w * 3 + 1] - qjs[jj * 3 + 1]));
      atomicAdd(&fAcc[w * 3 + 2], s * (qis[w * 3 + 2] - qjs[jj * 3 + 2]));
    }
  }
  __syncthreads();
  for (int idx = tid; idx < 48; idx += 512) atomicAdd(&fOut[ib * 3 + idx], fAcc[idx]);
}

// ---------------- Kernel E: KE reduction, dvdt, dpeta ----------------
__global__ void finalize_kernel(const float* __restrict__ v, const float* __restrict__ peta,
                                const float* __restrict__ mass, const float* __restrict__ fb,
                                float* __restrict__ out) {
  __shared__ float red[256];
  const int tid = threadIdx.x;
  float part = 0.0f;
  for (int idx = tid; idx < NATOMS * 3; idx += 256) {
    float vv = v[idx];
    part += mass[idx / 3] * vv * vv;
  }
  red[tid] = part;
  __syncthreads();
  for (int s = 128; s > 0; s >>= 1) {
    if (tid < s) red[tid] += red[tid + s];
    __syncthreads();
  }
  const float ke = 0.5f * red[0];
  const float Q0 = 2.0f * (float)(NATOMS * 3) * KT_C * TT2;
  const float pe0 = peta[0];
  for (int idx = tid; idx < NATOMS * 3; idx += 256) {
    float mi = mass[idx / 3];
    float p = mi * v[idx];
    out[idx] = (fb[idx] - pe0 * p / Q0) / mi;
    out[NATOMS * 3 + idx] = v[idx];
  }
  if (tid < 4) {
    const float Qc = 2.0f * KT_C * TT2;
    float p0 = peta[0], p1 = peta[1], p2 = peta[2], p3 = peta[3];
    float target = 0.5f * (float)(NATOMS * 3) * KT_C;
    float r;
    if (tid == 0)      r = 2.0f * (ke - target) - p0 * p1 / Qc;
    else if (tid == 1) r = p0 * p0 / Q0 - KT_C - p1 * p2 / Qc;
    else if (tid == 2) r = p1 * p1 / Qc - KT_C - p2 * p3 / Qc;
    else               r = p2 * p2 / Qc - KT_C;
    out[2 * NATOMS * 3 + tid] = r;
  }
}

extern "C" void kernel_launch(void* const* d_in, const int* in_sizes, int n_in,
                              void* d_out, int out_size, void* d_ws, size_t ws_size,
                              hipStream_t stream) {
  (void)in_sizes; (void)n_in; (void)out_size; (void)ws_size;
  const float* v     = (const float*)d_in[0];
  const float* q     = (const float*)d_in[1];
  const float* peta  = (const float*)d_in[2];
  const float* mass  = (const float*)d_in[3];
  const float* embed = (const float*)d_in[4];
  const float* W1    = (const float*)d_in[5];
  const float* b1    = (const float*)d_in[6];
  const float* W2    = (const float*)d_in[7];
  const float* b2    = (const float*)d_in[8];
  const float* W3    = (const float*)d_in[9];
  const float* w_out = (const float*)d_in[10];
  const int*   an    = (const int*)d_in[11];

  float* h0 = (float*)d_ws;
  float* mb = h0 + NATOMS * FEAT;
  float* ub = mb + NATOMS * FEAT;
  float* fb = ub + NATOMS * FEAT;

  init_kernel<<<(NATOMS * FEAT + 255) / 256, 256, 0, stream>>>(an, embed, h0, mb, fb);
  fwd_msg_kernel<<<dim3(NTILES, JCHUNKS), 512, 0, stream>>>(q, h0, W1, b1, W2, b2, mb);
  peratom_u_kernel<<<NATOMS, FEAT, 0, stream>>>(mb, W3, w_out, ub);
  bwd_force_kernel<<<dim3(NTILES, JCHUNKS), 512, 0, stream>>>(q, h0, ub, W1, b1, W2, fb);
  finalize_kernel<<<1, 256, 0, stream>>>(v, peta, mass, fb, (float*)d_out);
}